// KromHCHeadMixer_69166153335095
// MI455X (gfx1250) — compile-verified
//
#include <hip/hip_runtime.h>
#include <hip/hip_bf16.h>

// MI455X (gfx1250) fused head-mixer.
// Memory-bound (~285 MB moved, ~12 us at 23.3 TB/s). The one real GEMM
// (h = context @ W1^T, M=65536 N=96 K=64, f32) runs on V_WMMA_F32_16X16X4_F32.
// Output stores are non-temporal: out/H are write-once streams, and keeping
// them out of WGP$/L2 preserves x residency for the phase-6 re-read.

typedef float v2f __attribute__((ext_vector_type(2)));
typedef float v4f __attribute__((ext_vector_type(4)));
typedef float v8f __attribute__((ext_vector_type(8)));

#define BS   65536
#define NH   8
#define DD   64
#define MH   32
#define NK   3
#define WAVES 4
#define ROWS  16            // batch rows per wave (WMMA M-tile)
#define CTX_STRIDE 68       // floats; (4m+2h) mod 64 distinct -> conflict-free b64 frag loads
#define H_STRIDE   100      // floats; 4-aligned, 36L mod 64 distinct for 16 lanes
#define HV_STRIDE  12

__global__ __launch_bounds__(WAVES * 32)
void KromHCHeadMixer_kernel(const float* __restrict__ x,
                            const float* __restrict__ W1,
                            const float* __restrict__ W2,
                            float* __restrict__ out,
                            float* __restrict__ Hout)
{
    __shared__ float ctx [WAVES][ROWS * CTX_STRIDE]; // context, A-operand staging
    __shared__ float hbuf[WAVES][ROWS * H_STRIDE];   // h = relu(ctx @ W1^T), 16x96
    __shared__ float hv  [WAVES][ROWS * HV_STRIDE];  // 8 distinct H values per row

    const int lane = threadIdx.x & 31;
    const int wave = threadIdx.x >> 5;
    const int m16  = lane & 15;       // WMMA M/N lane index
    const int half = lane >> 4;       // WMMA K-half select
    const long b0  = (long)(blockIdx.x * WAVES + wave) * ROWS;

    const float* xw = x + b0 * (NH * DD);

    // ---- Phase 1: context = mean over heads. Lane t owns d = 2t..2t+1 (coalesced b64). ----
    for (int m = 0; m < ROWS; ++m) {
        const float* xr = xw + m * (NH * DD) + 2 * lane;
        v2f acc = {0.f, 0.f};
#pragma unroll
        for (int n = 0; n < NH; ++n) {
            v2f v = *(const v2f*)(xr + n * DD);
            acc.x += v.x; acc.y += v.y;
        }
        acc.x *= 0.125f; acc.y *= 0.125f;
        *(v2f*)&ctx[wave][m * CTX_STRIDE + 2 * lane] = acc;
    }
    __syncthreads();

    // ---- Phase 2: A-fragments for all 16 K-steps, loaded once, reused over 6 N-tiles.
    // v_wmma_f32_16x16x4_f32 A layout: v0 = A[m][4s+2h], v1 = A[m][4s+2h+1] -> one b64/step.
    v2f afrag[16];
#pragma unroll
    for (int s = 0; s < 16; ++s)
        afrag[s] = *(const v2f*)&ctx[wave][m16 * CTX_STRIDE + 4 * s + 2 * half];

    // ---- Phase 3: WMMA. B[d][col] = W1[col*64 + d] (col = k*32 + hh).
    // B frag (v0=B[4s+2h][col], v1=B[4s+2h+1][col]) is a contiguous float2 in W1 (L1-hot).
    for (int j = 0; j < 6; ++j) {
        const int col = j * 16 + m16;
        const float* bp = W1 + col * DD + 2 * half;
        v8f acc = {0.f, 0.f, 0.f, 0.f, 0.f, 0.f, 0.f, 0.f};
#pragma unroll
        for (int s = 0; s < 16; ++s) {
            v2f bfrag = *(const v2f*)(bp + 4 * s);
            acc = __builtin_amdgcn_wmma_f32_16x16x4_f32(
                      false, afrag[s], false, bfrag, (short)0, acc, false, false);
        }
        // D layout: VGPR r, lane L -> M = r + 8*(L>>4), N = 16j + (L&15)
#pragma unroll
        for (int r = 0; r < 8; ++r) {
            float hval = acc[r] > 0.f ? acc[r] : 0.f;   // relu
            hbuf[wave][(r + 8 * half) * H_STRIDE + col] = hval;
        }
    }
    __syncthreads();

    // ---- Phase 4: logits -> softmax -> Hv (8 distinct H entries: H[r][c] = Hv[r^c]). ----
    if (lane < 16) {
        const float* hrow = &hbuf[wave][m16 * H_STRIDE];
        float a0k[NK], a1k[NK];
#pragma unroll
        for (int k = 0; k < NK; ++k) {
            float l0 = 0.f, l1 = 0.f;
#pragma unroll
            for (int i = 0; i < MH; ++i) {
                float hh = hrow[k * MH + i];
                l0 += hh * W2[k * 2 * MH + i];        // broadcast (uniform addr) reads
                l1 += hh * W2[k * 2 * MH + MH + i];
            }
            float mx = fmaxf(l0, l1);
            float e0 = __expf(l0 - mx), e1 = __expf(l1 - mx);
            float inv = 1.f / (e0 + e1);
            a0k[k] = e0 * inv; a1k[k] = e1 * inv;
        }
#pragma unroll
        for (int t = 0; t < 8; ++t) {
            float v = ((t & 4) ? a1k[0] : a0k[0]) *
                      ((t & 2) ? a1k[1] : a0k[1]) *
                      ((t & 1) ? a1k[2] : a0k[2]);
            hv[wave][m16 * HV_STRIDE + t] = v;
        }
    }
    __syncthreads();

    // ---- Phase 5: coalesced H store (16 rows * 64 = 1024 floats contiguous), NT. ----
    float* hgl = Hout + b0 * 64;
#pragma unroll
    for (int it = 0; it < 8; ++it) {
        int g   = (it * 32 + lane) * 4;
        int row = g >> 6;
        int idx = g & 63;
        int r = idx >> 3, c = idx & 7;                // c is 0 or 4
        const float* hvr = &hv[wave][row * HV_STRIDE];
        v4f v4 = { hvr[r ^ c], hvr[r ^ (c + 1)], hvr[r ^ (c + 2)], hvr[r ^ (c + 3)] };
        __builtin_nontemporal_store(v4, (v4f*)(hgl + g));
    }

    // ---- Phase 6: out = H @ x. x re-read from global (WGP$/L2 hot). Lane t owns d=2t..2t+1.
    //      Streaming NT stores so out doesn't evict x. ----
    float* ow = out + b0 * (NH * DD);
    for (int m = 0; m < ROWS; ++m) {
        const float* xr = xw + m * (NH * DD) + 2 * lane;
        v2f xq[NH];
#pragma unroll
        for (int q = 0; q < NH; ++q) xq[q] = *(const v2f*)(xr + q * DD);
        const float* hvr = &hv[wave][m * HV_STRIDE];
#pragma unroll
        for (int n = 0; n < NH; ++n) {
            v2f acc = {0.f, 0.f};
#pragma unroll
            for (int q = 0; q < NH; ++q) {
                float hc = hvr[n ^ q];                 // LDS broadcast
                acc.x += hc * xq[q].x;
                acc.y += hc * xq[q].y;
            }
            __builtin_nontemporal_store(acc, (v2f*)(ow + m * (NH * DD) + n * DD + 2 * lane));
        }
    }
}

extern "C" void kernel_launch(void* const* d_in, const int* in_sizes, int n_in,
                              void* d_out, int out_size, void* d_ws, size_t ws_size,
                              hipStream_t stream) {
    const float* x  = (const float*)d_in[0];   // (65536, 8, 64)
    const float* W1 = (const float*)d_in[1];   // (3, 32, 64)
    const float* W2 = (const float*)d_in[2];   // (3, 2, 32)
    float* out  = (float*)d_out;                       // (65536, 8, 64) flat
    float* Hout = out + (size_t)BS * NH * DD;          // (65536, 8, 8) flat

    dim3 grid(BS / (WAVES * ROWS));   // 1024 blocks
    dim3 block(WAVES * 32);           // 128 threads = 4 waves
    KromHCHeadMixer_kernel<<<grid, block, 0, stream>>>(x, W1, W2, out, Hout);
}